// ScatLayerj1_20564303413876
// MI455X (gfx1250) — compile-verified
//
#include <hip/hip_runtime.h>
#include <math.h>

typedef __attribute__((ext_vector_type(2))) float v2f;
typedef __attribute__((ext_vector_type(8))) float v8f;

#define DN 8
#define DC 64
#define DH 256
#define DW 256
#define DH2 128
#define DW2 128

__device__ __forceinline__ int mir(int i, int S) {
  i = (i < 0) ? (-i - 1) : i;
  i = (i >= S) ? (2 * S - 1 - i) : i;
  return i;
}

// F0[m][k] = h0[d-1] for d=k-m in [1,5]   (5-tap lowpass)
__device__ __forceinline__ float sel_h0(int d, const float h0[5]) {
  float r = 0.0f;
  r = (d == 1) ? h0[0] : r;
  r = (d == 2) ? h0[1] : r;
  r = (d == 3) ? h0[2] : r;
  r = (d == 4) ? h0[3] : r;
  r = (d == 5) ? h0[4] : r;
  return r;
}
// F1[m][k] = h1[d] for d=k-m in [0,6]     (7-tap highpass)
__device__ __forceinline__ float sel_h1(int d, const float h1[7]) {
  float r = 0.0f;
  r = (d == 0) ? h1[0] : r;
  r = (d == 1) ? h1[1] : r;
  r = (d == 2) ? h1[2] : r;
  r = (d == 3) ? h1[3] : r;
  r = (d == 4) ? h1[4] : r;
  r = (d == 5) ? h1[5] : r;
  r = (d == 6) ? h1[6] : r;
  return r;
}

__device__ __forceinline__ float magf(float re, float im) {
  const float bias = 0.01f;
  return sqrtf(re * re + im * im + bias * bias) - bias;
}

__global__ __launch_bounds__(256) void dtcwt_j1_fused(
    const float* __restrict__ x, const float* __restrict__ h0o,
    const float* __restrict__ h1o, float* __restrict__ out) {
  // LDS: Lo/Hi row-filtered tiles (40 rows x 64 cols each) + a 32 KB region that
  // is first the x staging tile (40 x 72 pitch) and later the 4 column-filtered
  // full-res subband tiles (4 x 32 x 64). Plus two 40-entry zero-padded tap
  // tables for the banded WMMA A-operands. Total ~53.6 KB.
  __shared__ float sLo[40 * 64];
  __shared__ float sHi[40 * 64];
  __shared__ float sBuf[4 * 32 * 64];
  __shared__ float sT0[40];   // sT0[d+16] = F0 tap for band offset d=k-m
  __shared__ float sT1[40];   // sT1[d+16] = F1 tap for band offset d=k-m

  const int tid  = threadIdx.x;
  const int zz   = blockIdx.z;        // n*C + c
  const int row0 = blockIdx.y * 32;   // full-res tile origin (rows)
  const int col0 = blockIdx.x * 64;   // full-res tile origin (cols)

  float h0[5], h1[7];
#pragma unroll
  for (int i = 0; i < 5; ++i) h0[i] = h0o[i];
#pragma unroll
  for (int i = 0; i < 7; ++i) h1[i] = h1o[i];

  // ---- Phase 0: banded-matrix tap tables (one select chain, 40 threads, once).
  if (tid < 40) {
    const int d = tid - 16;
    sT0[tid] = sel_h0(d, h0);
    sT1[tid] = sel_h1(d, h1);
  }

  const float* xc = x + (size_t)zz * (DH * DW);

  // ---- Phase 1: stage x tile (rows row0-3 .. row0+36, cols col0-3 .. col0+66,
  //      symmetric-mirrored at image borders) into LDS.
  for (int idx = tid; idx < 40 * 70; idx += 256) {
    int l = idx / 70;
    int j = idx - l * 70;
    int gr = mir(row0 - 3 + l, DH);
    int gc = mir(col0 - 3 + j, DW);
    sBuf[l * 72 + j] = xc[gr * DW + gc];
  }
  __syncthreads();

  // ---- Phase 2: row filter (along W). Lo = 5-tap h0, Hi = 7-tap h1.
  for (int idx = tid; idx < 40 * 64; idx += 256) {
    int l = idx >> 6;
    int j = idx & 63;
    const float* xr = &sBuf[l * 72 + j];
    float v0 = xr[0], v1 = xr[1], v2 = xr[2], v3 = xr[3];
    float v4 = xr[4], v5 = xr[5], v6 = xr[6];
    sLo[idx] = h0[0] * v1 + h0[1] * v2 + h0[2] * v3 + h0[3] * v4 + h0[4] * v5;
    sHi[idx] = h1[0] * v0 + h1[1] * v1 + h1[2] * v2 + h1[3] * v3 +
               h1[4] * v4 + h1[5] * v5 + h1[6] * v6;
  }
  __syncthreads();

  // ---- Phase 3: column filter as banded matmul on the WMMA pipe.
  // Out[16x16] = F[16x24] x In[24x16]; K padded 22->24, tail coeffs are 0 and
  // rows 38/39 of sLo/sHi hold valid mirrored data so 0*finite stays 0.
  const int lane  = tid & 31;
  const int wave  = tid >> 5;
  const int m0    = (wave >> 2) << 4;  // 0 or 16: tile row block
  const int n0    = (wave & 3) << 4;   // 0/16/32/48: tile col block
  const int mm    = lane & 15;         // A: M row / B: N column
  const int khalf = (lane >> 4) << 1;  // 0 for lanes 0-15, 2 for lanes 16-31
  const int dbase = khalf - mm + 16;   // tap-table index for q=0, VGPR0

  v8f aLoLo = {0, 0, 0, 0, 0, 0, 0, 0};
  v8f aLoHi = {0, 0, 0, 0, 0, 0, 0, 0};
  v8f aHiLo = {0, 0, 0, 0, 0, 0, 0, 0};
  v8f aHiHi = {0, 0, 0, 0, 0, 0, 0, 0};

#pragma unroll
  for (int q = 0; q < 6; ++q) {
    const int kk = q * 4 + khalf;  // K index held by this lane (VGPR0), kk+1 in VGPR1
    v2f A0, A1, BLo, BHi;
    const int ti = q * 4 + dbase;  // = kk - mm + 16, in [1,38]
    A0.x = sT0[ti];
    A0.y = sT0[ti + 1];
    A1.x = sT1[ti];
    A1.y = sT1[ti + 1];
    const int l    = m0 + kk;      // window row (Lo/Hi LDS row = fullres row - (row0-3))
    const int colI = n0 + mm;
    BLo.x = sLo[l * 64 + colI];
    BLo.y = sLo[(l + 1) * 64 + colI];
    BHi.x = sHi[l * 64 + colI];
    BHi.y = sHi[(l + 1) * 64 + colI];
    aLoLo = __builtin_amdgcn_wmma_f32_16x16x4_f32(false, A0, false, BLo, (short)0, aLoLo, false, false);
    aLoHi = __builtin_amdgcn_wmma_f32_16x16x4_f32(false, A1, false, BLo, (short)0, aLoHi, false, false);
    aHiLo = __builtin_amdgcn_wmma_f32_16x16x4_f32(false, A0, false, BHi, (short)0, aHiLo, false, false);
    aHiHi = __builtin_amdgcn_wmma_f32_16x16x4_f32(false, A1, false, BHi, (short)0, aHiHi, false, false);
  }

  // ---- Phase 4: spill D tiles to LDS (C/D layout: N = lane&15, M = v + 8*(lane>>4)).
  {
    const int nloc  = lane & 15;
    const int mbase = (lane >> 4) << 3;
#pragma unroll
    for (int v = 0; v < 8; ++v) {
      const int r  = m0 + mbase + v;  // 0..31 full-res row within tile
      const int cc = n0 + nloc;       // 0..63 full-res col within tile
      sBuf[0 * 2048 + r * 64 + cc] = aLoLo[v];
      sBuf[1 * 2048 + r * 64 + cc] = aLoHi[v];
      sBuf[2 * 2048 + r * 64 + cc] = aHiLo[v];
      sBuf[3 * 2048 + r * 64 + cc] = aHiHi[v];
    }
  }
  __syncthreads();

  // ---- Phase 5: 2x2 pooling (LoLo), q2c + magnitude (6 oriented bands), store.
  const float sq = 0.70710678118654752440f;  // 1/sqrt(2)
  const int n = zz / DC;
  const int c = zz - n * DC;
  const size_t plane   = (size_t)DH2 * DW2;
  const size_t jstride = (size_t)DC * plane;
  float* ob = out + ((size_t)n * 7 * DC + c) * plane +
              (size_t)(row0 >> 1) * DW2 + (col0 >> 1);

  for (int idx = tid; idx < 16 * 32; idx += 256) {
    const int h2 = idx >> 5;
    const int w2 = idx & 31;
    const int r  = h2 << 1;
    const int cc = w2 << 1;
    float* o = ob + (size_t)h2 * DW2 + w2;

    // LoLo -> j=0 (2x2 mean)
    {
      const float* t = &sBuf[0 * 2048 + r * 64 + cc];
      o[0] = 0.25f * (t[0] + t[1] + t[64] + t[65]);
    }
    // LoHi -> j=1 (15 deg) and j=6 (165 deg)
    {
      const float* t = &sBuf[1 * 2048 + r * 64 + cc];
      float a = t[0], b = t[1], c2 = t[64], d = t[65];
      o[jstride * 1] = magf((a - d) * sq, (b + c2) * sq);
      o[jstride * 6] = magf((a + d) * sq, (b - c2) * sq);
    }
    // HiHi -> j=2 (45 deg) and j=5 (135 deg)
    {
      const float* t = &sBuf[3 * 2048 + r * 64 + cc];
      float a = t[0], b = t[1], c2 = t[64], d = t[65];
      o[jstride * 2] = magf((a - d) * sq, (b + c2) * sq);
      o[jstride * 5] = magf((a + d) * sq, (b - c2) * sq);
    }
    // HiLo -> j=3 (75 deg) and j=4 (105 deg)
    {
      const float* t = &sBuf[2 * 2048 + r * 64 + cc];
      float a = t[0], b = t[1], c2 = t[64], d = t[65];
      o[jstride * 3] = magf((a - d) * sq, (b + c2) * sq);
      o[jstride * 4] = magf((a + d) * sq, (b - c2) * sq);
    }
  }
}

extern "C" void kernel_launch(void* const* d_in, const int* in_sizes, int n_in,
                              void* d_out, int out_size, void* d_ws, size_t ws_size,
                              hipStream_t stream) {
  (void)in_sizes; (void)n_in; (void)out_size; (void)d_ws; (void)ws_size;
  const float* x   = (const float*)d_in[0];
  const float* h0o = (const float*)d_in[1];
  const float* h1o = (const float*)d_in[2];
  float* out = (float*)d_out;

  dim3 grid(DW / 64, DH / 32, DN * DC);  // (4, 8, 512)
  dim3 block(256);
  hipLaunchKernelGGL(dtcwt_j1_fused, grid, block, 0, stream, x, h0o, h1o, out);
}